// CopyAttention_54786602827854
// MI455X (gfx1250) — compile-verified
//
#include <hip/hip_runtime.h>
#include <hip/hip_bf16.h>
#include <math.h>

typedef __attribute__((ext_vector_type(2))) float v2f;
typedef __attribute__((ext_vector_type(8))) float v8f;

// Problem shapes (fixed by reference setup_inputs)
#define S_DIM 256
#define T_DIM 64
#define B_DIM 8
#define H_DIM 768
#define V_DIM 50257

__device__ __forceinline__ float fast_tanh(float x) {
#if __has_builtin(__builtin_amdgcn_tanhf)
    return __builtin_amdgcn_tanhf(x);   // V_TANH_F32 hardware transcendental
#else
    return tanhf(x);
#endif
}

// ---------------------------------------------------------------------------
// FP32 WMMA GEMM: C[M,N] = A[M,K] @ B[K,N] (+ bias[N]), all row-major fp32.
// One wave (32 threads) computes a 64x16 strip (4 M-tiles), reusing each B
// fragment across 4 V_WMMA_F32_16X16X4_F32 ops. M must be a multiple of 64.
// Out-of-range N columns: B column index is clamped (garbage contributions
// land only in accumulator lanes whose column is never stored).
// ---------------------------------------------------------------------------
__global__ __launch_bounds__(32) void wmma_gemm_f32_kernel(
    const float* __restrict__ A, const float* __restrict__ Bm,
    const float* __restrict__ bias, float* __restrict__ C,
    int M, int N, int K)
{
    const int ntile = blockIdx.x * 16;
    const int mbase = blockIdx.y * 64;      // 4 m-tiles of 16
    const int lane  = threadIdx.x;          // 0..31 (wave32)
    const int half  = lane >> 4;            // 0 or 1
    const int l16   = lane & 15;

    const int  n       = ntile + l16;                 // column this lane owns
    const bool ncol_ok = (n < N);
    const int  ncl     = ncol_ok ? n : (N - 1);       // clamped load column

    const float* __restrict__ a0 = A + (size_t)(mbase +  0 + l16) * K;
    const float* __restrict__ a1 = A + (size_t)(mbase + 16 + l16) * K;
    const float* __restrict__ a2 = A + (size_t)(mbase + 32 + l16) * K;
    const float* __restrict__ a3 = A + (size_t)(mbase + 48 + l16) * K;
    const float* __restrict__ bcol = Bm + ncl;

    v8f c0 = {}, c1 = {}, c2 = {}, c3 = {};
    for (int k0 = 0; k0 < K; k0 += 4) {
        const int ka = k0 + 2 * half;                 // this lane's K pair
        v2f b;                                        // B fragment (shared by 4 WMMAs)
        b.x = bcol[(size_t)ka * N];
        b.y = bcol[(size_t)(ka + 1) * N];
        v2f x0 = { a0[ka], a0[ka + 1] };              // contiguous 64-bit A loads
        v2f x1 = { a1[ka], a1[ka + 1] };
        v2f x2 = { a2[ka], a2[ka + 1] };
        v2f x3 = { a3[ka], a3[ka + 1] };
        c0 = __builtin_amdgcn_wmma_f32_16x16x4_f32(false, x0, false, b, (short)0, c0, false, false);
        c1 = __builtin_amdgcn_wmma_f32_16x16x4_f32(false, x1, false, b, (short)0, c1, false, false);
        c2 = __builtin_amdgcn_wmma_f32_16x16x4_f32(false, x2, false, b, (short)0, c2, false, false);
        c3 = __builtin_amdgcn_wmma_f32_16x16x4_f32(false, x3, false, b, (short)0, c3, false, false);
    }

    if (ncol_ok) {
        const float bv = bias ? bias[n] : 0.0f;
        float* __restrict__ cp = C + (size_t)(mbase + 8 * half) * N + n;
        #pragma unroll
        for (int r = 0; r < 8; ++r) cp[(size_t)(r     ) * N] = c0[r] + bv;
        #pragma unroll
        for (int r = 0; r < 8; ++r) cp[(size_t)(r + 16) * N] = c1[r] + bv;
        #pragma unroll
        for (int r = 0; r < 8; ++r) cp[(size_t)(r + 32) * N] = c2[r] + bv;
        #pragma unroll
        for (int r = 0; r < 8; ++r) cp[(size_t)(r + 48) * N] = c3[r] + bv;
    }
}

// ---------------------------------------------------------------------------
// attention[t,s,b] = sum_h tanh(dec_proj[t,b,h] + enc_proj[s,b,h]) * v[h] + b_v
// One wave per output scalar; lanes own float4 chunks, shuffle-reduce.
// ---------------------------------------------------------------------------
__global__ __launch_bounds__(256) void energy_kernel(
    const float* __restrict__ dec_proj, const float* __restrict__ enc_proj,
    const float* __restrict__ v_attn, const float* __restrict__ b_v,
    float* __restrict__ attn)
{
    const int wave = (blockIdx.x * blockDim.x + threadIdx.x) >> 5;
    const int lane = threadIdx.x & 31;
    const int total = T_DIM * S_DIM * B_DIM;
    if (wave >= total) return;

    const int b  = wave % B_DIM;
    const int ts = wave / B_DIM;
    const int s  = ts % S_DIM;
    const int t  = ts / S_DIM;

    const float* __restrict__ dp = dec_proj + ((size_t)t * B_DIM + b) * H_DIM;
    const float* __restrict__ ep = enc_proj + ((size_t)s * B_DIM + b) * H_DIM;

    float acc = 0.0f;
    for (int h = lane * 4; h < H_DIM; h += 128) {
        const float4 d = *(const float4*)(dp + h);
        const float4 e = *(const float4*)(ep + h);
        const float4 v = *(const float4*)(v_attn + h);
        acc += fast_tanh(d.x + e.x) * v.x;
        acc += fast_tanh(d.y + e.y) * v.y;
        acc += fast_tanh(d.z + e.z) * v.z;
        acc += fast_tanh(d.w + e.w) * v.w;
    }

    for (int off = 16; off > 0; off >>= 1)
        acc += __shfl_down(acc, off, 32);

    if (lane == 0) attn[wave] = acc + b_v[0];
}

// ---------------------------------------------------------------------------
// Masked softmax over S for each (t,b). attn layout (T,S,B), in place.
// ---------------------------------------------------------------------------
__global__ __launch_bounds__(S_DIM) void attn_softmax_kernel(
    float* __restrict__ attn, const int* __restrict__ src_ids)
{
    const int row = blockIdx.x;         // t*B + b
    const int t = row / B_DIM, b = row % B_DIM;
    const int s = threadIdx.x;          // S_DIM threads

    float val = attn[((size_t)t * S_DIM + s) * B_DIM + b];
    if (src_ids[(size_t)s * B_DIM + b] == 0) val = -1.0e10f;  // PAD mask

    __shared__ float red[S_DIM];
    red[s] = val; __syncthreads();
    for (int off = S_DIM / 2; off > 0; off >>= 1) {
        if (s < off) red[s] = fmaxf(red[s], red[s + off]);
        __syncthreads();
    }
    const float mx = red[0]; __syncthreads();

    const float e = __expf(val - mx);
    red[s] = e; __syncthreads();
    for (int off = S_DIM / 2; off > 0; off >>= 1) {
        if (s < off) red[s] += red[s + off];
        __syncthreads();
    }
    attn[((size_t)t * S_DIM + s) * B_DIM + b] = e / red[0];
}

// ---------------------------------------------------------------------------
// ctx[t,b,h] = sum_s attn_prob[t,s,b] * enc[s,b,h]
// ---------------------------------------------------------------------------
__global__ __launch_bounds__(256) void ctx_kernel(
    const float* __restrict__ attn_prob, const float* __restrict__ enc,
    float* __restrict__ ctx)
{
    const int i = blockIdx.x * blockDim.x + threadIdx.x;   // over T*B*H
    if (i >= T_DIM * B_DIM * H_DIM) return;
    const int h  = i % H_DIM;
    const int tb = i / H_DIM;
    const int b  = tb % B_DIM;
    const int t  = tb / B_DIM;

    float acc = 0.0f;
    for (int s = 0; s < S_DIM; ++s)
        acc += attn_prob[((size_t)t * S_DIM + s) * B_DIM + b]
             * enc[((size_t)s * B_DIM + b) * H_DIM + h];
    ctx[i] = acc;
}

// ---------------------------------------------------------------------------
// p_gen[t*B+b] = sigmoid([dec|ctx|tgt] . W_pgen + b_pgen). One wave per row.
// ---------------------------------------------------------------------------
__global__ __launch_bounds__(256) void pgen_kernel(
    const float* __restrict__ dec, const float* __restrict__ ctx,
    const float* __restrict__ tgt, const float* __restrict__ W_pgen,
    const float* __restrict__ b_pgen, float* __restrict__ p_gen)
{
    const int wave = (blockIdx.x * blockDim.x + threadIdx.x) >> 5;
    const int lane = threadIdx.x & 31;
    if (wave >= T_DIM * B_DIM) return;

    const float* __restrict__ d = dec + (size_t)wave * H_DIM;
    const float* __restrict__ c = ctx + (size_t)wave * H_DIM;
    const float* __restrict__ g = tgt + (size_t)wave * H_DIM;

    float acc = 0.0f;
    for (int h = lane; h < H_DIM; h += 32)
        acc += d[h] * W_pgen[h] + c[h] * W_pgen[H_DIM + h] + g[h] * W_pgen[2 * H_DIM + h];

    for (int off = 16; off > 0; off >>= 1)
        acc += __shfl_down(acc, off, 32);

    if (lane == 0)
        p_gen[wave] = 1.0f / (1.0f + __expf(-(acc + b_pgen[0])));
}

// ---------------------------------------------------------------------------
// In-place row softmax over V, scaled by p_gen[row]. One block per (t,b) row.
// ---------------------------------------------------------------------------
__global__ __launch_bounds__(256) void vocab_softmax_kernel(
    float* __restrict__ out, const float* __restrict__ p_gen)
{
    const int row = blockIdx.x;                 // t*B + b
    float* __restrict__ rp = out + (size_t)row * V_DIM;
    const int tid = threadIdx.x;

    __shared__ float red[256];

    float mx = -3.4e38f;
    for (int i = tid; i < V_DIM; i += 256) mx = fmaxf(mx, rp[i]);
    red[tid] = mx; __syncthreads();
    for (int off = 128; off > 0; off >>= 1) {
        if (tid < off) red[tid] = fmaxf(red[tid], red[tid + off]);
        __syncthreads();
    }
    mx = red[0]; __syncthreads();

    float sum = 0.0f;
    for (int i = tid; i < V_DIM; i += 256) sum += __expf(rp[i] - mx);
    red[tid] = sum; __syncthreads();
    for (int off = 128; off > 0; off >>= 1) {
        if (tid < off) red[tid] += red[tid + off];
        __syncthreads();
    }
    const float scale = p_gen[row] / red[0];

    for (int i = tid; i < V_DIM; i += 256) rp[i] = __expf(rp[i] - mx) * scale;
}

// ---------------------------------------------------------------------------
// final_prob[t,b,src_ids[s,b]] += attn_prob[t,s,b] * (1 - p_gen[t,b])
// ---------------------------------------------------------------------------
__global__ __launch_bounds__(256) void scatter_kernel(
    float* __restrict__ out, const float* __restrict__ attn_prob,
    const float* __restrict__ p_gen, const int* __restrict__ src_ids)
{
    const int i = blockIdx.x * blockDim.x + threadIdx.x;   // over T*S*B
    if (i >= T_DIM * S_DIM * B_DIM) return;
    const int b  = i % B_DIM;
    const int ts = i / B_DIM;
    const int s  = ts % S_DIM;
    const int t  = ts / S_DIM;

    const int idx  = src_ids[(size_t)s * B_DIM + b];
    const float ad = attn_prob[i] * (1.0f - p_gen[t * B_DIM + b]);
    atomicAdd(&out[((size_t)(t * B_DIM + b)) * V_DIM + idx], ad);
}

// ---------------------------------------------------------------------------
extern "C" void kernel_launch(void* const* d_in, const int* in_sizes, int n_in,
                              void* d_out, int out_size, void* d_ws, size_t ws_size,
                              hipStream_t stream) {
    const float* encoder_out   = (const float*)d_in[0];   // (S,B,H)
    const float* decoder_state = (const float*)d_in[1];   // (T,B,H)
    const int*   src_ids       = (const int*)  d_in[2];   // (S,B)
    const float* tgt_embedding = (const float*)d_in[3];   // (T,B,H)
    const float* W_attn        = (const float*)d_in[4];   // (2H,H)
    const float* b_attn        = (const float*)d_in[5];   // (H)
    const float* v_attn        = (const float*)d_in[6];   // (H)
    const float* b_v           = (const float*)d_in[7];   // (1)
    const float* W_pgen        = (const float*)d_in[8];   // (3H,1)
    const float* b_pgen        = (const float*)d_in[9];   // (1)
    const float* W_gen         = (const float*)d_in[10];  // (H,V)
    const float* b_gen         = (const float*)d_in[11];  // (V)
    float* out = (float*)d_out;                           // (T,B,V)

    // Workspace layout (floats)
    float* ws = (float*)d_ws;
    float* dec_proj  = ws;                                         // T*B*H
    float* enc_proj  = dec_proj + (size_t)T_DIM * B_DIM * H_DIM;   // S*B*H
    float* attn      = enc_proj + (size_t)S_DIM * B_DIM * H_DIM;   // T*S*B
    float* ctx       = attn     + (size_t)T_DIM * S_DIM * B_DIM;   // T*B*H
    float* p_gen     = ctx      + (size_t)T_DIM * B_DIM * H_DIM;   // T*B

    const int MB = T_DIM * B_DIM;        // 512 decoder rows
    const int ME = S_DIM * B_DIM;        // 2048 encoder rows

    // 1) dec_proj = decoder_state @ W_attn[:H]            (512 x 768)
    wmma_gemm_f32_kernel<<<dim3(H_DIM / 16, MB / 64), 32, 0, stream>>>(
        decoder_state, W_attn, nullptr, dec_proj, MB, H_DIM, H_DIM);

    // 2) enc_proj = encoder_out @ W_attn[H:] + b_attn     (2048 x 768)
    wmma_gemm_f32_kernel<<<dim3(H_DIM / 16, ME / 64), 32, 0, stream>>>(
        encoder_out, W_attn + (size_t)H_DIM * H_DIM, b_attn, enc_proj, ME, H_DIM, H_DIM);

    // 3) energy + v-dot -> attention (T,S,B)
    {
        const int waves = T_DIM * S_DIM * B_DIM;           // 131072
        const int blocks = (waves * 32 + 255) / 256;
        energy_kernel<<<blocks, 256, 0, stream>>>(dec_proj, enc_proj, v_attn, b_v, attn);
    }

    // 4) masked softmax over S, in place
    attn_softmax_kernel<<<MB, S_DIM, 0, stream>>>(attn, src_ids);

    // 5) ctx = einsum('tsb,sbh->tbh')
    {
        const int n = T_DIM * B_DIM * H_DIM;
        ctx_kernel<<<(n + 255) / 256, 256, 0, stream>>>(attn, encoder_out, ctx);
    }

    // 6) p_gen
    {
        const int blocks = (MB * 32 + 255) / 256;
        pgen_kernel<<<blocks, 256, 0, stream>>>(decoder_state, ctx, tgt_embedding,
                                                W_pgen, b_pgen, p_gen);
    }

    // 7) logits = decoder_state @ W_gen + b_gen, written directly into d_out
    {
        const int ntiles = (V_DIM + 15) / 16;              // 3142
        wmma_gemm_f32_kernel<<<dim3(ntiles, MB / 64), 32, 0, stream>>>(
            decoder_state, W_gen, b_gen, out, MB, V_DIM, H_DIM);
    }

    // 8) softmax over V, scaled by p_gen (in place on d_out)
    vocab_softmax_kernel<<<MB, 256, 0, stream>>>(out, p_gen);

    // 9) scatter-add copy distribution
    {
        const int n = T_DIM * S_DIM * B_DIM;
        scatter_kernel<<<(n + 255) / 256, 256, 0, stream>>>(out, attn, p_gen, src_ids);
    }
}